// DemonsRegistration_15006615733533
// MI455X (gfx1250) — compile-verified
//
#include <hip/hip_runtime.h>

// Demons registration, 96^3, 50 iterations — fused per-voxel recurrence.
// vel update is broadcast to all 3 channels from zero init => velocity is ONE
// scalar field u per voxel with no spatial coupling => all 50 iterations run
// in a single kernel. moving (3.5 MB) stays resident in the 192 MB L2 for the
// 8-corner gathers. The kernel is L2-latency bound (serial dependence through
// u), so each thread runs TWO independent recurrences (z and z+48) for 2x
// memory-level parallelism. Block tiles (fixed, grad; both z-slabs) are staged
// into LDS with a single 3D Tensor Data Mover descriptor each
// (tensor_load_to_lds + s_wait_tensorcnt).

#define DIMN 96
#define NVOX (96 * 96 * 96)          // 884736
#define SLICE (96 * 96)              // 9216
#define NITERS 50
#define EPSV 1e-8f

// Block covers a 32(x) x 8(y) tile in TWO z-slices (z, z+48): 256 threads.
#define TX 32
#define TY 8
#define GX 3                          // 96/32
#define GY 12                         // 96/8
#define GZ 48                         // two slabs per block: z and z+48
#define ZSTEP 48
#define NBLOCKS (GX * GY * GZ)        // 1728

typedef unsigned int v4u __attribute__((ext_vector_type(4)));
typedef int          v8i __attribute__((ext_vector_type(8)));
typedef int          v4i __attribute__((ext_vector_type(4)));

#define HAVE_TDM (__has_builtin(__builtin_amdgcn_tensor_load_to_lds) && \
                  __has_builtin(__builtin_amdgcn_s_wait_tensorcnt))

__device__ __forceinline__ int clampi(int v, int lo, int hi) {
    return v < lo ? lo : (v > hi ? hi : v);
}

// Trilinear sample, align_corners=False / padding_mode='zeros'.
// px,py,pz in pixel space: p = 96*i/95 - 0.5 + 48*u.
__device__ __forceinline__ float sample_tri(const float* __restrict__ mv,
                                            float px, float py, float pz) {
    float fx0 = floorf(px), fy0 = floorf(py), fz0 = floorf(pz);
    int ix = (int)fx0, iy = (int)fy0, iz = (int)fz0;
    float tx = px - fx0, ty = py - fy0, tz = pz - fz0;

    // Zero weights of out-of-range corners (zeros padding); clamp indices so
    // the loads themselves are always in-bounds.
    float wx0 = ((unsigned)ix       < 96u) ? (1.0f - tx) : 0.0f;
    float wx1 = ((unsigned)(ix + 1) < 96u) ? tx          : 0.0f;
    float wy0 = ((unsigned)iy       < 96u) ? (1.0f - ty) : 0.0f;
    float wy1 = ((unsigned)(iy + 1) < 96u) ? ty          : 0.0f;
    float wz0 = ((unsigned)iz       < 96u) ? (1.0f - tz) : 0.0f;
    float wz1 = ((unsigned)(iz + 1) < 96u) ? tz          : 0.0f;

    int cx0 = clampi(ix,     0, 95);
    int cx1 = clampi(ix + 1, 0, 95);
    int cy0 = clampi(iy,     0, 95) * DIMN;
    int cy1 = clampi(iy + 1, 0, 95) * DIMN;
    int cz0 = clampi(iz,     0, 95) * SLICE;
    int cz1 = clampi(iz + 1, 0, 95) * SLICE;

    const float* p00 = mv + (cz0 + cy0);
    const float* p01 = mv + (cz0 + cy1);
    const float* p10 = mv + (cz1 + cy0);
    const float* p11 = mv + (cz1 + cy1);

    float s00 = fmaf(wx1, p00[cx1], wx0 * p00[cx0]);
    float s01 = fmaf(wx1, p01[cx1], wx0 * p01[cx0]);
    float s10 = fmaf(wx1, p10[cx1], wx0 * p10[cx0]);
    float s11 = fmaf(wx1, p11[cx1], wx0 * p11[cx0]);

    float s0 = fmaf(wy1, s01, wy0 * s00);
    float s1 = fmaf(wy1, s11, wy0 * s10);
    return fmaf(wz1, s1, wz0 * s0);
}

#if HAVE_TDM
// One TDM 3D tile load: TX x TY x 2 tile (two z-slabs, ZSTEP apart) -> LDS.
// D# per CDNA5 ISA §8.3-8.6:
//   group0: count=1 | lds_addr | global_addr[56:0] | type=2
//   group1: data_size=4B, tensor_dim0/1=96, tile=(32,8,2),
//           dim0_stride=96, dim1_stride=ZSTEP*SLICE (z-slab step)
//   group2: tensor_dim2=96, tile_dim3=0      group3: unused
// Builtin on this toolchain is the 6-arg form (trailing v8i zero-filled).
__device__ __forceinline__ void tdm_load_tile2(const float* gsrc_tile,
                                               unsigned lds_byte_off) {
    unsigned long long ga = (unsigned long long)(uintptr_t)gsrc_tile;
    v4u g0;
    g0[0] = 1u;                                   // count=1, user descriptor
    g0[1] = lds_byte_off;                         // lds_addr (bytes)
    g0[2] = (unsigned)(ga & 0xFFFFFFFFull);       // global_addr[31:0]
    g0[3] = (unsigned)((ga >> 32) & 0x01FFFFFFull) | (2u << 30); // [56:32]|type=2

    const unsigned zstride = (unsigned)(ZSTEP * SLICE);   // 442368 elems
    v8i g1;
    g1[0] = (int)(2u << 16);                      // data_size=2 (4B), no wg mask
    g1[1] = (int)((unsigned)DIMN << 16);          // tensor_dim0[15:0]=96
    g1[2] = (int)((unsigned)DIMN << 16);          // dim0[31:16]=0 | tensor_dim1[15:0]=96
    g1[3] = (int)((unsigned)TX << 16);            // dim1[31:16]=0 | tile_dim0=32
    g1[4] = (int)(TY | (2u << 16));               // tile_dim1=8 | tile_dim2=2
    g1[5] = DIMN;                                 // tensor_dim0_stride=96
    g1[6] = (int)((zstride & 0xFFFFu) << 16);     // dim0_str[47:32]=0 | dim1_str[15:0]
    g1[7] = (int)(zstride >> 16);                 // tensor_dim1_stride[47:16]

    v4i g2 = {DIMN, 0, 0, 0};                     // tensor_dim2=96; tile_dim3=0
    v4i g3 = {0, 0, 0, 0};
    v8i gz8 = {0, 0, 0, 0, 0, 0, 0, 0};
    __builtin_amdgcn_tensor_load_to_lds(g0, g1, g2, g3, gz8, 0);
}
#endif

// Kernel A: fixed-image "gradient" with the reference's triple-overwrite
// semantics: x-central-diff if x interior, else y, else z, else 0.
__global__ void k_grad(const float* __restrict__ fx, float* __restrict__ g) {
    int x = blockIdx.x * TX + threadIdx.x;
    int y = blockIdx.y * TY + threadIdx.y;
    int z = blockIdx.z;
    int v = (z * DIMN + y) * DIMN + x;
    float gv = 0.0f;
    if (x >= 1 && x <= 94)      gv = (fx[v + 1]     - fx[v - 1])     * 0.5f;
    else if (y >= 1 && y <= 94) gv = (fx[v + DIMN]  - fx[v - DIMN])  * 0.5f;
    else if (z >= 1 && z <= 94) gv = (fx[v + SLICE] - fx[v - SLICE]) * 0.5f;
    g[v] = gv;
}

// Kernel B: TDM-stage fixed/grad tiles (both slabs) to LDS, run TWO fused
// 50-iteration recurrences per thread, final warp, outputs, MSE partials.
__global__ void k_demons(const float* __restrict__ mv,
                         const float* __restrict__ fx,
                         const float* __restrict__ g,
                         float* __restrict__ out,
                         float* __restrict__ partials) {
    __shared__ float s_fx[2][TY][TX];
    __shared__ float s_g[2][TY][TX];
    __shared__ float sred[TX * TY];

    int x  = blockIdx.x * TX + threadIdx.x;
    int y  = blockIdx.y * TY + threadIdx.y;
    int z0 = blockIdx.z;                 // slab 0
    int z1 = z0 + ZSTEP;                 // slab 1
    int v0 = (z0 * DIMN + y) * DIMN + x;
    int v1 = v0 + ZSTEP * SLICE;
    int tid = threadIdx.y * TX + threadIdx.x;

    int tile_base = (z0 * DIMN + (int)(blockIdx.y * TY)) * DIMN
                  + (int)(blockIdx.x * TX);

#if HAVE_TDM
    // Wave 0 issues both 3D tile DMAs (TDM ignores EXEC; once per wave).
    if (threadIdx.y == 0) {
        tdm_load_tile2(fx + tile_base, (unsigned)(uintptr_t)&s_fx[0][0][0]);
        tdm_load_tile2(g  + tile_base, (unsigned)(uintptr_t)&s_g[0][0][0]);
        __builtin_amdgcn_s_wait_tensorcnt(0);
    }
    __syncthreads();
#else
    s_fx[0][threadIdx.y][threadIdx.x] = fx[tile_base + threadIdx.y * DIMN + threadIdx.x];
    s_g [0][threadIdx.y][threadIdx.x] = g [tile_base + threadIdx.y * DIMN + threadIdx.x];
    s_fx[1][threadIdx.y][threadIdx.x] = fx[tile_base + ZSTEP * SLICE + threadIdx.y * DIMN + threadIdx.x];
    s_g [1][threadIdx.y][threadIdx.x] = g [tile_base + ZSTEP * SLICE + threadIdx.y * DIMN + threadIdx.x];
    __syncthreads();
#endif

    // Pixel-space bases: ((linspace(-1,1,96)[i]+1)*96 - 1)*0.5 = 96*i/95 - 0.5.
    const float scale = 96.0f / 95.0f;
    float bx  = fmaf((float)x,  scale, -0.5f);
    float by  = fmaf((float)y,  scale, -0.5f);
    float bz0 = fmaf((float)z0, scale, -0.5f);
    float bz1 = fmaf((float)z1, scale, -0.5f);

    float fxv0 = s_fx[0][threadIdx.y][threadIdx.x];
    float fxv1 = s_fx[1][threadIdx.y][threadIdx.x];
    float gv0  = s_g[0][threadIdx.y][threadIdx.x];
    float gv1  = s_g[1][threadIdx.y][threadIdx.x];
    float g2e0 = fmaf(gv0, gv0, EPSV);            // g^2 + eps (ALPHA == 1)
    float g2e1 = fmaf(gv1, gv1, EPSV);

    // Warm the L2 gather path (gfx1250 global_prefetch_b8).
    __builtin_prefetch(mv + v0, 0, 0);
    __builtin_prefetch(mv + v1, 0, 0);

    float u0 = 0.0f, u1 = 0.0f;
#pragma unroll 1
    for (int it = 0; it < NITERS; ++it) {
        // Two independent recurrences: 16 gathers in flight per iteration.
        float o0 = 48.0f * u0;
        float o1 = 48.0f * u1;
        float w0 = sample_tri(mv, bx + o0, by + o0, bz0 + o0);
        float w1 = sample_tri(mv, bx + o1, by + o1, bz1 + o1);
        float d0 = fxv0 - w0;
        float d1 = fxv1 - w1;
        u0 += (d0 * gv0) / fmaf(d0, d0, g2e0);    // g^2 + diff^2 + eps
        u1 += (d1 * gv1) / fmaf(d1, d1, g2e1);
    }

    float o0 = 48.0f * u0;
    float o1 = 48.0f * u1;
    float wf0 = sample_tri(mv, bx + o0, by + o0, bz0 + o0);
    float wf1 = sample_tri(mv, bx + o1, by + o1, bz1 + o1);

    out[v0] = wf0;                                // final_warped
    out[v1] = wf1;
    out[NVOX + v0]     = u0;                      // velocity x/y/z (broadcast)
    out[2 * NVOX + v0] = u0;
    out[3 * NVOX + v0] = u0;
    out[NVOX + v1]     = u1;
    out[2 * NVOX + v1] = u1;
    out[3 * NVOX + v1] = u1;

    float e0 = wf0 - fxv0;
    float e1 = wf1 - fxv1;
    sred[tid] = fmaf(e0, e0, e1 * e1);
    __syncthreads();
#pragma unroll
    for (int s = (TX * TY) / 2; s > 0; s >>= 1) {
        if (tid < s) sred[tid] += sred[tid + s];
        __syncthreads();
    }
    if (tid == 0) {
        int bid = (int)blockIdx.x + GX * ((int)blockIdx.y + GY * (int)blockIdx.z);
        partials[bid] = sred[0];
    }
}

// Kernel C: deterministic final reduction of the 1728 partials -> mean.
__global__ void k_reduce(const float* __restrict__ partials,
                         float* __restrict__ out_mse) {
    __shared__ float sred[256];
    float acc = 0.0f;
    for (int i = threadIdx.x; i < NBLOCKS; i += 256) acc += partials[i];
    sred[threadIdx.x] = acc;
    __syncthreads();
#pragma unroll
    for (int s = 128; s > 0; s >>= 1) {
        if (threadIdx.x < s) sred[threadIdx.x] += sred[threadIdx.x + s];
        __syncthreads();
    }
    if (threadIdx.x == 0) out_mse[0] = sred[0] * (1.0f / (float)NVOX);
}

extern "C" void kernel_launch(void* const* d_in, const int* in_sizes, int n_in,
                              void* d_out, int out_size, void* d_ws, size_t ws_size,
                              hipStream_t stream) {
    (void)in_sizes; (void)n_in; (void)out_size; (void)ws_size;

    const float* moving = (const float*)d_in[0];
    const float* fixedp = (const float*)d_in[1];
    float* out = (float*)d_out;

    // Workspace: [ grad field: NVOX floats | partials: NBLOCKS floats ]
    float* g        = (float*)d_ws;
    float* partials = g + NVOX;

    dim3 blk(TX, TY, 1);
    k_grad  <<<dim3(GX, GY, DIMN), blk, 0, stream>>>(fixedp, g);
    k_demons<<<dim3(GX, GY, GZ),   blk, 0, stream>>>(moving, fixedp, g, out, partials);
    k_reduce<<<1, 256, 0, stream>>>(partials, out + 4 * NVOX);
}